// SimplifiedGCNConv_88313117541032
// MI455X (gfx1250) — compile-verified
//
#include <hip/hip_runtime.h>

typedef float v2f __attribute__((ext_vector_type(2)));
typedef float v4f __attribute__((ext_vector_type(4)));
typedef float v8f __attribute__((ext_vector_type(8)));

#define IN_CH 128
#define OUT_CH 128
#define LDS_STRIDE 132          // 128 + 4 pad: 528B rows, 16B aligned, conflict-free b64 reads
#define ROWS_PER_WAVE 16
#define ROWS_PER_BLOCK 128      // 8 waves * 16 rows

// ---------------------------------------------------------------------------
// Kernel 1: y = x @ W^T + b  via V_WMMA_F32_16X16X4_F32 (full f32 precision).
// Writes y to BOTH d_ws (gather source for scatter) and d_out (self term).
// ---------------------------------------------------------------------------
__global__ __launch_bounds__(256) void gcn_gemm_wmma(
    const float* __restrict__ x, const float* __restrict__ W,
    const float* __restrict__ b, float* __restrict__ y,
    float* __restrict__ out, int N)
{
  __shared__ float Wlds[OUT_CH * LDS_STRIDE];

  const int tid = threadIdx.x;

  // Stage W (128x128 f32 = 64KB) into padded LDS: 4096 float4 chunks / 256 thr.
#pragma unroll
  for (int i = 0; i < 16; ++i) {
    int idx = tid + i * 256;
    int r = idx >> 5;
    int c = (idx & 31) << 2;
    *(v4f*)&Wlds[r * LDS_STRIDE + c] = *(const v4f*)&W[r * IN_CH + c];
  }
  __syncthreads();

  const int wave = tid >> 5;
  const int lane = tid & 31;
  const int half = lane >> 4;      // 0: lanes 0-15, 1: lanes 16-31
  const int l16  = lane & 15;
  const int koff = half * 2;       // A/B fragment K offset per ISA f32 layout

  const int rb = blockIdx.x * ROWS_PER_BLOCK + wave * ROWS_PER_WAVE;

  // A fragments: 16x4 f32 tile per step; lane holds x[rb+l16][4s+koff .. +1].
  int arow = rb + l16;
  int arow_c = (arow < N) ? arow : (N - 1);   // clamp (stores predicated later)
  const float* xrow = x + (size_t)arow_c * IN_CH + koff;

  v2f afrag[32];
#pragma unroll
  for (int s = 0; s < 32; ++s)
    afrag[s] = *(const v2f*)(xrow + 4 * s);

  // 8 column tiles of 16 output channels each.
#pragma unroll
  for (int t = 0; t < 8; ++t) {
    const int ob = t * 16;

    // Bias preloaded: every acc element of this lane maps to column ob+l16.
    float bv = b[ob + l16];
    v8f acc = { bv, bv, bv, bv, bv, bv, bv, bv };

    // B fragment: B[k][n] = W[ob+n][k]; lane reads W[ob+l16][4s+koff .. +1].
    const float* wrow = &Wlds[(ob + l16) * LDS_STRIDE + koff];

#pragma unroll
    for (int s = 0; s < 32; ++s) {
      v2f bfrag = *(const v2f*)(wrow + 4 * s);
      acc = __builtin_amdgcn_wmma_f32_16x16x4_f32(
          /*neg_a=*/false, afrag[s], /*neg_b=*/false, bfrag,
          /*c_mod=*/(short)0, acc, /*reuse_a=*/false, /*reuse_b=*/false);
    }

    // C/D layout: VGPR j -> row (half*8 + j), lane%16 -> column.
#pragma unroll
    for (int j = 0; j < 8; ++j) {
      int row = rb + half * 8 + j;
      if (row < N) {
        size_t o = (size_t)row * OUT_CH + (ob + l16);
        y[o]   = acc[j];
        out[o] = acc[j];
      }
    }
  }
}

// ---------------------------------------------------------------------------
// Kernel 2: out[dst[e]][:] += y[src[e]][:]  — one wave per edge, lane handles
// 4 consecutive channels (coalesced b128 gather + 4x global_atomic_add_f32).
// ---------------------------------------------------------------------------
__global__ __launch_bounds__(256) void gcn_scatter_add(
    const int* __restrict__ ei, const float* __restrict__ y,
    float* __restrict__ out, int E)
{
  const int lane = threadIdx.x & 31;
  const int wid  = (int)((blockIdx.x * blockDim.x + threadIdx.x) >> 5);
  const int nw   = (int)((gridDim.x * blockDim.x) >> 5);

  for (int e = wid; e < E; e += nw) {
    int s = ei[e];        // wave-uniform
    int d = ei[E + e];    // wave-uniform

    v4f v = *(const v4f*)(y + (size_t)s * OUT_CH + lane * 4);
    float* o = out + (size_t)d * OUT_CH + lane * 4;
    unsafeAtomicAdd(o + 0, v.x);
    unsafeAtomicAdd(o + 1, v.y);
    unsafeAtomicAdd(o + 2, v.z);
    unsafeAtomicAdd(o + 3, v.w);
  }
}

// ---------------------------------------------------------------------------
extern "C" void kernel_launch(void* const* d_in, const int* in_sizes, int n_in,
                              void* d_out, int out_size, void* d_ws, size_t ws_size,
                              hipStream_t stream) {
  const float* x  = (const float*)d_in[0];
  const int*   ei = (const int*)d_in[1];   // edge_index [2, E], int
  const float* W  = (const float*)d_in[2];
  const float* b  = (const float*)d_in[3];
  float* out = (float*)d_out;
  float* y   = (float*)d_ws;               // [N, 128] scratch = x_transformed

  const int N = in_sizes[0] / IN_CH;
  const int E = in_sizes[1] / 2;

  const int grid1 = (N + ROWS_PER_BLOCK - 1) / ROWS_PER_BLOCK;
  gcn_gemm_wmma<<<grid1, 256, 0, stream>>>(x, W, b, y, out, N);

  const int grid2 = 2048;                  // 16384 edge-waves, grid-stride
  gcn_scatter_add<<<grid2, 256, 0, stream>>>(ei, y, out, E);
}